// EdgeEncoding_56530359550892
// MI455X (gfx1250) — compile-verified
//
#include <hip/hip_runtime.h>

typedef __attribute__((ext_vector_type(2))) float v2f;
typedef __attribute__((ext_vector_type(4))) float v4f;
typedef __attribute__((ext_vector_type(8))) float v8f;

#define NUM_HEADS 8
#define HIDDEN 16
#define WAVES_PER_BLOCK 8

// ---------------------------------------------------------------------------
// Kernel 1: zero the (N,N,8) fp32 output. 128 MiB of pure stores -> this is
// the bandwidth-bound part. Use b128 non-temporal stores so we stream at HBM
// rate without displacing useful L2 lines (only ~1.5% is later overwritten).
// ---------------------------------------------------------------------------
__global__ void zero_fill_kernel(v4f* __restrict__ out, long long n4,
                                 float* __restrict__ tail, int ntail) {
  long long i = (long long)blockIdx.x * blockDim.x + threadIdx.x;
  const long long stride = (long long)gridDim.x * blockDim.x;
  const v4f z = {0.f, 0.f, 0.f, 0.f};
  for (; i < n4; i += stride) {
    __builtin_nontemporal_store(z, &out[i]);
  }
  if (blockIdx.x == 0 && threadIdx.x < (unsigned)ntail) {
    tail[threadIdx.x] = 0.0f;
  }
}

// ---------------------------------------------------------------------------
// Kernel 2: edge MLP via fp32 WMMA (V_WMMA_F32_16X16X4_F32) + scatter.
// One wave32 processes a 16-edge tile:
//   GEMM1: A = edge_attr tile (16x3 padded to 16x4), B = W1 (3x16 pad 4x16),
//          C = b1 broadcast  -> one 16x16x4 WMMA.
//   ReLU on accumulators, round-trip through LDS (D-layout -> A-layout).
//   GEMM2: H(16x16) x W2(16x8 pad 16x16) = 4 chained K=4 WMMAs, C = b2.
//   Scatter rows to out[(src*nn+dst)*8 + head].
// ---------------------------------------------------------------------------
__global__ void edge_mlp_scatter_kernel(const int*   __restrict__ edge_index,
                                        const float* __restrict__ edge_attr,
                                        const int*   __restrict__ num_nodes_p,
                                        const float* __restrict__ W1,
                                        const float* __restrict__ b1,
                                        const float* __restrict__ W2,
                                        const float* __restrict__ b2,
                                        float* __restrict__ out,
                                        int E) {
  __shared__ float h_lds[WAVES_PER_BLOCK][16 * 16];  // per-wave 16x16 H tile
  __shared__ int   idx_lds[WAVES_PER_BLOCK][16];     // per-wave scatter bases

  const int lane    = threadIdx.x & 31;
  const int wave    = threadIdx.x >> 5;
  const int tile    = blockIdx.x * WAVES_PER_BLOCK + wave;
  const int e0      = tile * 16;                 // first edge of this tile
  const int halfSel = lane >> 4;                 // 0: lanes 0-15, 1: lanes 16-31
  const int lmod    = lane & 15;
  const int nn      = num_nodes_p[0];

  const bool active = (e0 < E);                  // wave-uniform

  v8f acc;
  if (active) {
    // ---- A matrix: 16 edges x K{f0,f1,f2,pad} (ISA 16x4 fp32 A layout) ----
    v2f a;
    {
      const int e = e0 + lmod;
      if (halfSel == 0) {
        a.x = edge_attr[3 * e + 0];              // K=0
        a.y = edge_attr[3 * e + 1];              // K=1
      } else {
        a.x = edge_attr[3 * e + 2];              // K=2
        a.y = 0.0f;                              // K=3 (pad)
      }
    }
    // ---- B matrix: W1 rows striped across lanes, K rows split by half ----
    v2f bm;
    if (halfSel == 0) {
      bm.x = W1[0 * HIDDEN + lmod];              // K=0 row
      bm.y = W1[1 * HIDDEN + lmod];              // K=1 row
    } else {
      bm.x = W1[2 * HIDDEN + lmod];              // K=2 row
      bm.y = 0.0f;                               // K=3 pad row
    }
    // ---- C: bias b1[n] broadcast over all 16 rows ----
    v8f c;
    const float bias1 = b1[lmod];
#pragma unroll
    for (int v = 0; v < 8; ++v) c[v] = bias1;

    v8f h = __builtin_amdgcn_wmma_f32_16x16x4_f32(
        /*neg_a=*/false, a, /*neg_b=*/false, bm,
        /*c_mod=*/(short)0, c, /*reuse_a=*/false, /*reuse_b=*/false);

    // ---- ReLU ----
#pragma unroll
    for (int v = 0; v < 8; ++v) h[v] = fmaxf(h[v], 0.0f);

    // ---- D-layout -> row-major LDS tile (M = v + 8*half, N = lmod) ----
#pragma unroll
    for (int v = 0; v < 8; ++v) {
      const int M = v + halfSel * 8;
      h_lds[wave][M * 16 + lmod] = h[v];
    }
    // ---- scatter base indices (one per edge row) ----
    if (halfSel == 0) {
      const int e = e0 + lmod;
      const int s = edge_index[e];
      const int d = edge_index[E + e];
      idx_lds[wave][lmod] = (s * nn + d) * NUM_HEADS;
    }
  }
  __syncthreads();

  if (active) {
    // ---- GEMM2: H(16x16) x W2(16x8 pad to 16x16), 4 chained K=4 WMMAs ----
    const float bias2 = (lmod < NUM_HEADS) ? b2[lmod] : 0.0f;
#pragma unroll
    for (int v = 0; v < 8; ++v) acc[v] = bias2;

#pragma unroll
    for (int kc = 0; kc < 4; ++kc) {
      const int kBase = kc * 4 + halfSel * 2;
      v2f a2, b2m;
      // A-layout: lane row M=lmod, two consecutive K values
      a2.x = h_lds[wave][lmod * 16 + kBase + 0];
      a2.y = h_lds[wave][lmod * 16 + kBase + 1];
      // B-layout: W2 rows kBase, kBase+1 striped over lanes (N pad >= 8)
      if (lmod < NUM_HEADS) {
        b2m.x = W2[(kBase + 0) * NUM_HEADS + lmod];
        b2m.y = W2[(kBase + 1) * NUM_HEADS + lmod];
      } else {
        b2m.x = 0.0f;
        b2m.y = 0.0f;
      }
      acc = __builtin_amdgcn_wmma_f32_16x16x4_f32(
          false, a2, false, b2m, (short)0, acc, false, false);
    }

    // ---- scatter: head-lane n writes 8 edge rows; 8 lanes x 4B contiguous ----
    if (lmod < NUM_HEADS) {
#pragma unroll
      for (int v = 0; v < 8; ++v) {
        const int M    = v + halfSel * 8;
        const int base = idx_lds[wave][M];
        out[base + lmod] = acc[v];
      }
    }
  }
}

extern "C" void kernel_launch(void* const* d_in, const int* in_sizes, int n_in,
                              void* d_out, int out_size, void* d_ws, size_t ws_size,
                              hipStream_t stream) {
  const int*   edge_index = (const int*)d_in[0];   // (2, E) int32
  const float* edge_attr  = (const float*)d_in[1]; // (E, 3) f32
  const int*   num_nodes  = (const int*)d_in[2];   // scalar int
  const float* W1         = (const float*)d_in[3]; // (3, 16)
  const float* b1         = (const float*)d_in[4]; // (16,)
  const float* W2         = (const float*)d_in[5]; // (16, 8)
  const float* b2         = (const float*)d_in[6]; // (8,)
  float*       out        = (float*)d_out;         // (N, N, 8) f32

  const int E = in_sizes[0] / 2;

  // --- 1) zero-fill output (bandwidth-bound: 128 MiB of NT b128 stores) ---
  const long long n4    = (long long)out_size >> 2;
  const int       ntail = out_size & 3;
  zero_fill_kernel<<<4096, 256, 0, stream>>>((v4f*)out, n4,
                                             out + (n4 << 2), ntail);

  // --- 2) MLP (fp32 WMMA) + scatter; 16 edges/wave, 8 waves/block ---
  const int edges_per_block = WAVES_PER_BLOCK * 16;
  const int blocks = (E + edges_per_block - 1) / edges_per_block;
  edge_mlp_scatter_kernel<<<blocks, 256, 0, stream>>>(
      edge_index, edge_attr, num_nodes, W1, b1, W2, b2, out, E);
}